// AssetScoringHead_39857296507195
// MI455X (gfx1250) — compile-verified
//
#include <hip/hip_runtime.h>
#include <math.h>

// ---------------------------------------------------------------------------
// AssetScoringHead on MI455X (gfx1250, wave32).
//   B=64, N_ASSETS=4096, D=1024, fp32.
//   sW = S @ Wb                 [64,1024]   tiled WMMA f32 16x16x4 + async LDS
//   hs = S @ W1[:d]             [64,1024]   tiled WMMA f32 16x16x4 + async LDS
//   hb = E @ W1[d:] + b1        [4096,1024] tiled WMMA f32 16x16x4 + async LDS
//   score[b,n] = dot(sW[b],E[n]) + sum_k gelu(hs[b,k]+hb[n,k])*w2[k] + bb + b2
//   out = softmax(score, axis=-1)
// h [64,4096,1024] is never materialized; hb (16MB) lives in the 192MB L2.
// All GEMM K-loops are double-buffered through LDS with
// global_load_async_to_lds_b128 (ASYNCcnt) so WMMAs never wait on HBM/L2.
// ---------------------------------------------------------------------------

#define D_MODEL   1024
#define N_ASSETS  4096
#define BATCH     64

typedef float v2f __attribute__((ext_vector_type(2)));
typedef float v8f __attribute__((ext_vector_type(8)));
typedef int   v4i __attribute__((ext_vector_type(4)));

#if defined(__has_builtin)
#if __has_builtin(__builtin_amdgcn_global_load_async_to_lds_b128)
#define ASYNC_LDS 1
#endif
#endif

__device__ __forceinline__ void wait_async_all() {
#ifdef ASYNC_LDS
#if __has_builtin(__builtin_amdgcn_s_wait_asynccnt)
  __builtin_amdgcn_s_wait_asynccnt(0);
#else
  asm volatile("s_wait_asynccnt 0x0" ::: "memory");
#endif
#endif
}

__device__ __forceinline__ void copy16_to_lds(const float* src, float* lds_dst) {
#ifdef ASYNC_LDS
  // Async DMA global->LDS, tracked by ASYNCcnt; no VGPR round-trip.
  __builtin_amdgcn_global_load_async_to_lds_b128(
      (__attribute__((address_space(1))) v4i*)src,
      (__attribute__((address_space(3))) v4i*)lds_dst, 0, 0);
#else
  *(float4*)lds_dst = *(const float4*)src;
#endif
}

__device__ __forceinline__ float gelu_exact(float x) {
  // torch nn.GELU default (exact erf form)
  return 0.5f * x * (1.0f + erff(x * 0.70710678118654752f));
}

// ---------------------------------------------------------------------------
// Tiled fp32 WMMA GEMM (N = K = 1024 hardcoded): C[M,1024] = A[M,1024] @ B
// (+ optional bias[N] broadcast over rows).
// Block = WAVES waves; each wave owns 16 M-rows x 64 N-cols (4 v8f accums).
// K consumed in 32-wide stages, double-buffered in LDS via async DMA:
//   prefetch stage kt+1 (async, other buffer) -> 32 WMMAs from LDS (cur)
//   -> s_wait_asynccnt 0 -> barrier.
// LDS strides: A rows 36 dwords (conflict-free ds_load_b64 fragments, 16B
// aligned rows for b128 async writes); B rows 80 dwords (the two K-halves of
// a fragment hit disjoint bank groups: bank = 16*r + c mod 64).
// ---------------------------------------------------------------------------
#define BN 64
#define BK 32
#define NSTAGES (D_MODEL / BK)
#define AS_STRIDE 36
#define BS_STRIDE 80

template <int BM, int WAVES>
__global__ __launch_bounds__(WAVES * 32)
void wmma_gemm_tiled_kernel(const float* __restrict__ A,    // [M,1024]
                            const float* __restrict__ B,    // [1024,1024]
                            const float* __restrict__ bias, // [1024] or null
                            float* __restrict__ C)          // [M,1024]
{
  static_assert(BM == WAVES * 16, "one wave per 16 M-rows");
  __shared__ float As[2][BM * AS_STRIDE];
  __shared__ float Bs[2][BK * BS_STRIDE];

  const int tid  = threadIdx.x;
  const int lane = tid & 31;
  const int wave = tid >> 5;        // M sub-tile (16 rows each)
  const int row  = lane & 15;
  const int hi   = lane >> 4;

  const int m0 = blockIdx.x * BM;
  const int n0 = blockIdx.y * BN;   // gridDim.y = 16

  auto stage_copy = [&](int k0, int buf) {
    // A tile: BM x 32 f32 = BM*8 float4 -> 4 per thread
    #pragma unroll
    for (int i = 0; i < 4; ++i) {
      const int f  = tid + WAVES * 32 * i;
      const int ra = f >> 3;
      const int c4 = (f & 7) << 2;
      copy16_to_lds(A + (size_t)(m0 + ra) * D_MODEL + k0 + c4,
                    &As[buf][ra * AS_STRIDE + c4]);
    }
    // B tile: 32 x 64 f32 = 512 float4
    #pragma unroll
    for (int f = tid; f < 512; f += WAVES * 32) {
      const int rb = f >> 4;
      const int c4 = (f & 15) << 2;
      copy16_to_lds(B + (size_t)(k0 + rb) * D_MODEL + n0 + c4,
                    &Bs[buf][rb * BS_STRIDE + c4]);
    }
  };

  v8f acc[4];
  #pragma unroll
  for (int j = 0; j < 4; ++j) {
    const float bv = bias ? bias[n0 + 16 * j + row] : 0.0f;  // C(m,n) init
    #pragma unroll
    for (int v = 0; v < 8; ++v) acc[j][v] = bv;
  }

  const int abase = (wave * 16 + row) * AS_STRIDE + 2 * hi;

  stage_copy(0, 0);
  wait_async_all();
  __syncthreads();

  for (int kt = 0; kt < NSTAGES; ++kt) {
    const int buf = kt & 1;
    if (kt + 1 < NSTAGES) stage_copy((kt + 1) * BK, buf ^ 1);  // prefetch

    #pragma unroll
    for (int ks = 0; ks < BK; ks += 4) {
      v2f a;
      a.x = As[buf][abase + ks];       // ds_load_b64
      a.y = As[buf][abase + ks + 1];
      #pragma unroll
      for (int j = 0; j < 4; ++j) {
        const int bb0 = (ks + 2 * hi) * BS_STRIDE + 16 * j + row;
        v2f bf;
        bf.x = Bs[buf][bb0];
        bf.y = Bs[buf][bb0 + BS_STRIDE];
        acc[j] = __builtin_amdgcn_wmma_f32_16x16x4_f32(
            false, a, false, bf, (short)0, acc[j], false, false);
      }
    }

    if (kt + 1 < NSTAGES) {
      wait_async_all();   // next buffer filled (own DMA done) ...
      __syncthreads();    // ... and everyone is done reading cur buffer
    }
  }

  #pragma unroll
  for (int j = 0; j < 4; ++j) {
    float* __restrict__ Cp = C + (size_t)(m0 + 8 * hi) * D_MODEL + n0 + 16 * j + row;
    #pragma unroll
    for (int v = 0; v < 8; ++v) Cp[(size_t)v * D_MODEL] = acc[j][v];
  }
}

// ---------------------------------------------------------------------------
// Fused score kernel: one wave per (asset n, 16-row batch tile).
// Lanes stride K (coalesced 128B loads); hb & E rows reused across 16 batch
// rows -> ~130MB HBM total; the 268M exact-gelu (erf) evals dominate (VALU).
// ---------------------------------------------------------------------------
__global__ __launch_bounds__(256)
void score_kernel(const float* __restrict__ sW,
                  const float* __restrict__ hs,
                  const float* __restrict__ hb,
                  const float* __restrict__ asset_emb,
                  const float* __restrict__ w2,
                  const float* __restrict__ bilinear_b,
                  const float* __restrict__ b2,
                  float* __restrict__ scores)
{
  const int lane = threadIdx.x & 31;
  const int wave = threadIdx.x >> 5;
  const int gw   = blockIdx.x * 8 + wave;      // 16384 waves total
  const int n    = gw >> 2;                    // asset index
  const int b0   = (gw & 3) * 16;              // batch tile base

  float accb[16], accm[16];
  #pragma unroll
  for (int i = 0; i < 16; ++i) { accb[i] = 0.0f; accm[i] = 0.0f; }

  const float* __restrict__ hbp = hb + (size_t)n * D_MODEL;
  const float* __restrict__ aep = asset_emb + (size_t)n * D_MODEL;

  for (int k0 = 0; k0 < D_MODEL; k0 += 32) {
    const int k = k0 + lane;
    const float hbv = hbp[k];
    const float aev = aep[k];
    const float w2v = w2[k];
    #pragma unroll
    for (int bb = 0; bb < 16; ++bb) {
      const float hsv = hs[(size_t)(b0 + bb) * D_MODEL + k];
      const float swv = sW[(size_t)(b0 + bb) * D_MODEL + k];
      accb[bb] = fmaf(swv, aev, accb[bb]);
      accm[bb] = fmaf(w2v, gelu_exact(hsv + hbv), accm[bb]);
    }
  }

  const float cbias = bilinear_b[0] + b2[0];
  #pragma unroll
  for (int bb = 0; bb < 16; ++bb) {
    float v = accb[bb] + accm[bb];
    #pragma unroll
    for (int off = 16; off > 0; off >>= 1) v += __shfl_xor(v, off, 32);
    if (lane == 0) scores[(size_t)(b0 + bb) * N_ASSETS + n] = v + cbias;
  }
}

// ---------------------------------------------------------------------------
// Row softmax: one 256-thread block per batch row (4096 columns).
// ---------------------------------------------------------------------------
__global__ __launch_bounds__(256)
void softmax_kernel(const float* __restrict__ scores, float* __restrict__ out)
{
  __shared__ float red[256];
  const int b = blockIdx.x;
  const int t = threadIdx.x;
  const float* __restrict__ srow = scores + (size_t)b * N_ASSETS;
  float* __restrict__ orow = out + (size_t)b * N_ASSETS;

  float m = -3.402823466e38f;
  for (int n = t; n < N_ASSETS; n += 256) m = fmaxf(m, srow[n]);
  red[t] = m; __syncthreads();
  for (int s = 128; s > 0; s >>= 1) {
    if (t < s) red[t] = fmaxf(red[t], red[t + s]);
    __syncthreads();
  }
  m = red[0]; __syncthreads();

  float sum = 0.0f;
  for (int n = t; n < N_ASSETS; n += 256) {
    const float e = expf(srow[n] - m);
    orow[n] = e;
    sum += e;
  }
  red[t] = sum; __syncthreads();
  for (int s = 128; s > 0; s >>= 1) {
    if (t < s) red[t] += red[t + s];
    __syncthreads();
  }
  const float inv = 1.0f / red[0];
  __syncthreads();
  for (int n = t; n < N_ASSETS; n += 256) orow[n] *= inv;
}

// ---------------------------------------------------------------------------
extern "C" void kernel_launch(void* const* d_in, const int* in_sizes, int n_in,
                              void* d_out, int out_size, void* d_ws, size_t ws_size,
                              hipStream_t stream) {
  const float* market_state = (const float*)d_in[0];   // [64,1024]
  const float* asset_emb    = (const float*)d_in[1];   // [4096,1024]
  const float* bilinear_w   = (const float*)d_in[2];   // [1024,1024]
  const float* bilinear_b   = (const float*)d_in[3];   // [1]
  const float* w1           = (const float*)d_in[4];   // [2048,1024]
  const float* b1           = (const float*)d_in[5];   // [1024]
  const float* w2           = (const float*)d_in[6];   // [1024,1]
  const float* b2           = (const float*)d_in[7];   // [1]
  float* out = (float*)d_out;

  // Workspace layout (floats): sW | hs | hb | scores  (~17.5 MB)
  float* ws     = (float*)d_ws;
  float* sW     = ws;                                   // 64*1024
  float* hsbuf  = ws + BATCH * D_MODEL;                 // 64*1024
  float* hb     = hsbuf + BATCH * D_MODEL;              // 4096*1024
  float* scores = hb + (size_t)N_ASSETS * D_MODEL;      // 64*4096

  // sW = S @ Wb : M=64 -> BM=64, 4 waves, grid (1,16)
  wmma_gemm_tiled_kernel<64, 4><<<dim3(1, 16), 128, 0, stream>>>(
      market_state, bilinear_w, nullptr, sW);
  // hs = S @ W1[:d]
  wmma_gemm_tiled_kernel<64, 4><<<dim3(1, 16), 128, 0, stream>>>(
      market_state, w1, nullptr, hsbuf);

  // hb = E @ W1[d:] + b1 : M=4096 -> BM=128, 8 waves, grid (32,16)
  wmma_gemm_tiled_kernel<128, 8><<<dim3(32, 16), 256, 0, stream>>>(
      asset_emb, w1 + (size_t)D_MODEL * D_MODEL, b1, hb);

  // Fused bilinear dot + gelu-MLP contraction: 16384 waves -> 2048 blocks
  score_kernel<<<2048, 256, 0, stream>>>(
      sW, hsbuf, hb, asset_emb, w2, bilinear_b, b2, scores);

  // Softmax over assets per batch row
  softmax_kernel<<<BATCH, 256, 0, stream>>>(scores, out);
}